// ModuleDSepconvOptimized_44547400794795
// MI455X (gfx1250) — compile-verified
//
#include <hip/hip_runtime.h>
#include <stdint.h>

#define GLOBAL_AS __attribute__((address_space(1)))
#define LDS_AS    __attribute__((address_space(3)))

typedef float v4f __attribute__((ext_vector_type(4)));

namespace {
constexpr int Bb = 2, Cc = 3, Hh = 512, Ww = 512, Ff = 5, Tt = 25;
constexpr int HW = Hh * Ww;
constexpr int TW = 32, TH = 8, RAD = 8;
constexpr int PW = TW + 2 * RAD;   // 48
constexpr int PH = TH + 2 * RAD;   // 24
constexpr int NPIX = PW * PH;      // 1152 halo pixels
constexpr int NTHREADS = TW * TH;  // 256 = 8 wave32
}

// gfx1250 async copy: global -> LDS, tracked by ASYNCcnt (no VGPR round trip)
__device__ __forceinline__ void async_f32_to_lds(const float* gsrc, void* ldst) {
#if __has_builtin(__builtin_amdgcn_global_load_async_to_lds_b32)
  __builtin_amdgcn_global_load_async_to_lds_b32(
      (GLOBAL_AS int*)(uintptr_t)gsrc,
      (LDS_AS int*)(uintptr_t)ldst, /*offset=*/0, /*cpol=*/0);
#else
  asm volatile("global_load_async_to_lds_b32 %0, %1, off"
               :: "v"((uint32_t)(uintptr_t)ldst), "v"((uint64_t)(uintptr_t)gsrc)
               : "memory");
#endif
}

__device__ __forceinline__ void wait_async_all() {
#if __has_builtin(__builtin_amdgcn_s_wait_asynccnt)
  __builtin_amdgcn_s_wait_asynccnt(0);
#else
  asm volatile("s_wait_asynccnt 0" ::: "memory");
#endif
}

__global__ __launch_bounds__(NTHREADS) void dsepconv_kernel(
    const float* __restrict__ inp,  const float* __restrict__ vert,
    const float* __restrict__ horz, const float* __restrict__ offx,
    const float* __restrict__ offy, const float* __restrict__ mask,
    float* __restrict__ out) {
  __shared__ v4f tile[NPIX];  // channel-interleaved (c in .x/.y/.z), 18 KB

  const int tx = threadIdx.x, ty = threadIdx.y;
  const int tid = ty * TW + tx;
  const int b = blockIdx.z;
  const int x0t = blockIdx.x * TW - RAD;   // tile origin incl. halo
  const int y0t = blockIdx.y * TH - RAD;

  const float* inpB = inp + b * (Cc * HW);

  // Stage halo tile with clamped source coords via async-to-LDS.
  for (int i = tid; i < NPIX; i += NTHREADS) {
    const int lyy = i / PW, lxx = i - lyy * PW;
    const int gy = min(max(y0t + lyy, 0), Hh - 1);
    const int gx = min(max(x0t + lxx, 0), Ww - 1);
    const float* src = inpB + gy * Ww + gx;
    char* dst = (char*)&tile[i];
    async_f32_to_lds(src,          dst);       // c=0 -> .x
    async_f32_to_lds(src + HW,     dst + 4);   // c=1 -> .y
    async_f32_to_lds(src + 2 * HW, dst + 8);   // c=2 -> .z
  }

  const int w = blockIdx.x * TW + tx;
  const int h = blockIdx.y * TH + ty;
  const int p = h * Ww + w;

  // Coefficient preloads overlap the async tile fill.
  float vv[Ff], hh[Ff];
  const float* vP = vert + b * (Ff * HW) + p;
  const float* hP = horz + b * (Ff * HW) + p;
#pragma unroll
  for (int f = 0; f < Ff; ++f) {
    vv[f] = __builtin_nontemporal_load(vP + f * HW);
    hh[f] = __builtin_nontemporal_load(hP + f * HW);
  }
  const float* mP  = mask + b * (Tt * HW) + p;
  const float* oxP = offx + b * (Tt * HW) + p;
  const float* oyP = offy + b * (Tt * HW) + p;

  wait_async_all();
  __syncthreads();

  // Opaque LDS base: low 32 bits of the flat address of `tile` are its LDS
  // byte offset (ISA aperture rule). Laundering it through empty asm stops
  // the optimizer from value-folding the tile loads (it cannot prove what
  // the async-to-LDS DMA wrote); the &~15 re-establishes 16B alignment so
  // the backend selects ds_load_b128.
  uint32_t tbase = (uint32_t)(uintptr_t)(void*)tile;
  asm volatile("" : "+v"(tbase));
  tbase &= ~15u;
  const LDS_AS v4f* tl = (const LDS_AS v4f*)(uintptr_t)tbase;

  float acc0 = 0.f, acc1 = 0.f, acc2 = 0.f;

#pragma unroll
  for (int fy = 0; fy < Ff; ++fy) {
    // iy_unclamped = ox + h + fy - c_off + 1 - 0.5 (normalize/denorm collapsed)
    const float by = (float)(h + fy) - 1.5f;
    const float vcoef = vv[fy];
#pragma unroll
    for (int fx = 0; fx < Ff; ++fx) {
      const int t = fy * Ff + fx;
      const float bx = (float)(w + fx) - 1.5f;
      const float m  = __builtin_nontemporal_load(mP + t * HW);
      const float ox = __builtin_nontemporal_load(oxP + t * HW);  // y coord (faithful swap)
      const float oy = __builtin_nontemporal_load(oyP + t * HW);  // x coord (faithful swap)
      const float coef = vcoef * hh[fx] * m;

      float ix = fminf(fmaxf(oy + bx, -0.5f), (float)Ww - 0.5f);
      float iy = fminf(fmaxf(ox + by, -0.5f), (float)Hh - 0.5f);

      const float xf = floorf(ix), yf = floorf(iy);
      const float wx1 = ix - xf, wy1 = iy - yf;
      const float wx0 = 1.f - wx1, wy0 = 1.f - wy1;

      const int x0 = (int)xf, y0 = (int)yf;
      const int x0i = min(max(x0, 0), Ww - 1);
      const int x1i = min(max(x0 + 1, 0), Ww - 1);
      const int y0i = min(max(y0, 0), Hh - 1);
      const int y1i = min(max(y0 + 1, 0), Hh - 1);

      const int lx0 = x0i - x0t, lx1 = x1i - x0t;
      const int ly0 = y0i - y0t, ly1 = y1i - y0t;

      float a0, a1, a2, e0, e1, e2, c0, c1, c2, d0, d1, d2;
      if ((((lx0 | ly0) >= 0) & (lx1 < PW)) & (ly1 < PH)) {
        // Fast path: 4x ds_load_b128 from the staged tile.
        const v4f q00 = tl[ly0 * PW + lx0];
        const v4f q01 = tl[ly0 * PW + lx1];
        const v4f q10 = tl[ly1 * PW + lx0];
        const v4f q11 = tl[ly1 * PW + lx1];
        a0 = q00.x; a1 = q00.y; a2 = q00.z;
        e0 = q01.x; e1 = q01.y; e2 = q01.z;
        c0 = q10.x; c1 = q10.y; c2 = q10.z;
        d0 = q11.x; d1 = q11.y; d2 = q11.z;
      } else {
        // Rare outlier offset: correct global gather (L2-resident image).
        const float* r0 = inpB + y0i * Ww;
        const float* r1 = inpB + y1i * Ww;
        a0 = r0[x0i]; a1 = r0[HW + x0i]; a2 = r0[2 * HW + x0i];
        e0 = r0[x1i]; e1 = r0[HW + x1i]; e2 = r0[2 * HW + x1i];
        c0 = r1[x0i]; c1 = r1[HW + x0i]; c2 = r1[2 * HW + x0i];
        d0 = r1[x1i]; d1 = r1[HW + x1i]; d2 = r1[2 * HW + x1i];
      }

      const float w00 = coef * (wy0 * wx0);
      const float w01 = coef * (wy0 * wx1);
      const float w10 = coef * (wy1 * wx0);
      const float w11 = coef * (wy1 * wx1);
      acc0 += w00 * a0 + w01 * e0 + w10 * c0 + w11 * d0;
      acc1 += w00 * a1 + w01 * e1 + w10 * c1 + w11 * d1;
      acc2 += w00 * a2 + w01 * e2 + w10 * c2 + w11 * d2;
    }
  }

  float* oP = out + b * (Cc * HW) + p;
  __builtin_nontemporal_store(acc0, oP);
  __builtin_nontemporal_store(acc1, oP + HW);
  __builtin_nontemporal_store(acc2, oP + 2 * HW);
}

extern "C" void kernel_launch(void* const* d_in, const int* in_sizes, int n_in,
                              void* d_out, int out_size, void* d_ws, size_t ws_size,
                              hipStream_t stream) {
  (void)in_sizes; (void)n_in; (void)out_size; (void)d_ws; (void)ws_size;
  const float* inp  = (const float*)d_in[0];
  const float* vert = (const float*)d_in[1];
  const float* horz = (const float*)d_in[2];
  const float* offx = (const float*)d_in[3];
  const float* offy = (const float*)d_in[4];
  const float* mask = (const float*)d_in[5];
  dim3 grid(Ww / TW, Hh / TH, Bb);
  dim3 block(TW, TH, 1);
  dsepconv_kernel<<<grid, block, 0, stream>>>(inp, vert, horz, offx, offy, mask,
                                              (float*)d_out);
}